// Model_39676907887843
// MI455X (gfx1250) — compile-verified
//
#include <hip/hip_runtime.h>
#include <hip/hip_bf16.h>

// out = x1 @ W^11  (collapsed 11-matmul chain, shared 20x20 weight)
//
// Phase 1: W_eff = W^11 in fp32, pre-swizzled into V_WMMA_F32_16X16X4_F32
//          B-fragment layout (2 N-tiles x 5 K-chunks x 32 lanes x float2) in d_ws.
// Phase 2: streaming 2M x 20 GEMM via fp32 WMMA; HBM-bound (320 MB total).

typedef __attribute__((ext_vector_type(2))) float v2f;
typedef __attribute__((ext_vector_type(8))) float v8f;

#define N_ROWS   2000000
#define SIZE     20
#define CHAIN    11
#define TILES    (N_ROWS / 16)        // 125000
#define TILES_PER_WAVE 5
#define WAVES_PER_BLOCK 8
#define GRID_BLOCKS (TILES / (TILES_PER_WAVE * WAVES_PER_BLOCK))  // 3125

// ---------------------------------------------------------------------------
// Phase 1: one block. Compute W^11 in LDS, then emit WMMA B fragments to ws.
// Fragment f = tile*5 + chunk; per lane l: float2 {V0,V1} at ws[f*64 + l*2].
//   l <  16: V0 = Weff[4c+0][16t + l],    V1 = Weff[4c+1][16t + l]
//   l >= 16: V0 = Weff[4c+2][16t + l-16], V1 = Weff[4c+3][16t + l-16]
// (cols >= 20 padded with zero for the second N-tile)
// ---------------------------------------------------------------------------
__global__ void weight_pow_frag_kernel(const float* __restrict__ W,
                                       float* __restrict__ ws) {
    __shared__ float base[SIZE * SIZE];
    __shared__ float cur[SIZE * SIZE];
    __shared__ float nxt[SIZE * SIZE];

    const int tid = threadIdx.x;
    for (int i = tid; i < SIZE * SIZE; i += blockDim.x) {
        float w = W[i];
        base[i] = w;
        cur[i]  = w;   // W^1
    }
    __syncthreads();

    for (int it = 0; it < CHAIN - 1; ++it) {          // -> W^11
        for (int e = tid; e < SIZE * SIZE; e += blockDim.x) {
            const int i = e / SIZE, j = e % SIZE;
            float s = 0.0f;
#pragma unroll
            for (int k = 0; k < SIZE; ++k)
                s = fmaf(cur[i * SIZE + k], base[k * SIZE + j], s);
            nxt[e] = s;
        }
        __syncthreads();
        for (int e = tid; e < SIZE * SIZE; e += blockDim.x) cur[e] = nxt[e];
        __syncthreads();
    }

    // Emit B fragments (10 frags x 32 lanes), one thread per (frag,lane).
    if (tid < 10 * 32) {
        const int f = tid >> 5;          // 0..9
        const int l = tid & 31;          // lane
        const int t = f / 5;             // N-tile (0: cols 0-15, 1: cols 16-31)
        const int c = f % 5;             // K-chunk (K = 4c..4c+3)
        const int col  = t * 16 + (l & 15);
        const int krow = 4 * c + ((l >= 16) ? 2 : 0);
        float v0 = 0.0f, v1 = 0.0f;
        if (col < SIZE) {
            v0 = cur[(krow + 0) * SIZE + col];
            v1 = cur[(krow + 1) * SIZE + col];
        }
        ws[f * 64 + l * 2 + 0] = v0;
        ws[f * 64 + l * 2 + 1] = v1;
    }
}

// ---------------------------------------------------------------------------
// Phase 2: out = x1 @ W_eff using V_WMMA_F32_16X16X4_F32.
// One wave owns 5 consecutive 16-row tiles; 2 N-tiles x 5 K-chunks per tile.
// ---------------------------------------------------------------------------
__global__ void __launch_bounds__(256) chain_gemm_kernel(
    const float* __restrict__ x,
    const float* __restrict__ wsB,
    float* __restrict__ out) {

    const int lane = threadIdx.x & 31;
    const int wave = blockIdx.x * WAVES_PER_BLOCK + (threadIdx.x >> 5);

    // B fragments: resident in registers for all 5 tiles (L1-cached loads).
    v2f b[2][5];
#pragma unroll
    for (int t = 0; t < 2; ++t)
#pragma unroll
        for (int c = 0; c < 5; ++c)
            b[t][c] = *(const v2f*)(wsB + ((t * 5 + c) * 64 + lane * 2));

    const int rowInTile = lane & 15;
    const int ko        = (lane >> 4) << 1;   // 0 (lanes 0-15) or 2 (lanes 16-31)
    const int mBase     = (lane >> 4) << 3;   // C/D: 0 or 8
    const int colStore  = lane & 15;

    const long tile0 = (long)wave * TILES_PER_WAVE;

    for (int tt = 0; tt < TILES_PER_WAVE; ++tt) {
        const long r = (tile0 + tt) * 16;

        // A fragments: each lane reads its half-row as 5 float2's.
        const float* rp = x + (r + rowInTile) * SIZE + ko;
        v2f a[5];
#pragma unroll
        for (int c = 0; c < 5; ++c)
            a[c] = *(const v2f*)(rp + 4 * c);

        v8f acc0 = {};
        v8f acc1 = {};
#pragma unroll
        for (int c = 0; c < 5; ++c) {
            // (neg_a, A, neg_b, B, c_mod, C, reuse_a, reuse_b)
            acc0 = __builtin_amdgcn_wmma_f32_16x16x4_f32(
                false, a[c], false, b[0][c], (short)0, acc0, false, false);
            acc1 = __builtin_amdgcn_wmma_f32_16x16x4_f32(
                false, a[c], false, b[1][c], (short)0, acc1, false, false);
        }

        // Store tile 0 (cols 0-15): VGPR v -> row r+mBase+v, col = colStore.
        float* op = out + (r + mBase) * SIZE + colStore;
#pragma unroll
        for (int v = 0; v < 8; ++v)
            op[v * SIZE] = acc0[v];

        // Store tile 1 (cols 16-19 only).
        if (colStore < SIZE - 16) {
            float* op1 = out + (r + mBase) * SIZE + 16 + colStore;
#pragma unroll
            for (int v = 0; v < 8; ++v)
                op1[v * SIZE] = acc1[v];
        }
    }
}

extern "C" void kernel_launch(void* const* d_in, const int* in_sizes, int n_in,
                              void* d_out, int out_size, void* d_ws, size_t ws_size,
                              hipStream_t stream) {
    const float* x1 = (const float*)d_in[0];   // [2,000,000 x 20] fp32
    const float* W  = (const float*)d_in[1];   // [20 x 20] fp32
    float* out = (float*)d_out;                // [2,000,000 x 20] fp32
    float* ws  = (float*)d_ws;                 // needs 640 floats (2560 B)

    weight_pow_frag_kernel<<<1, 320, 0, stream>>>(W, ws);
    chain_gemm_kernel<<<GRID_BLOCKS, 256, 0, stream>>>(x1, ws, out);
}